// CartesianToJacobi_29334626632267
// MI455X (gfx1250) — compile-verified
//
#include <hip/hip_runtime.h>

typedef __attribute__((ext_vector_type(2))) float v2f;
typedef __attribute__((ext_vector_type(8))) float v8f;

static __device__ __forceinline__ v8f wmma4(v2f a, v2f b, v8f c) {
  // D = A(16x4 f32) * B(4x16 f32) + C(16x16 f32), exact f32 math
  return __builtin_amdgcn_wmma_f32_16x16x4_f32(
      /*neg_a=*/false, a, /*neg_b=*/false, b,
      /*c_mod=*/(short)0, c, /*reuse_a=*/false, /*reuse_b=*/false);
}

// One wave (32 lanes) per batch. 256 elements viewed as X[r][c] = x[16r+c].
// Pass 1: Y = X * U  (U upper-tri ones)  -> per-row inclusive scan (4 WMMAs)
// Pass 2: S = Lstrict * bcast(T) + Y     -> add row carries        (4 WMMAs)
// S = full 256-element inclusive scan, in the 16x16 f32 C/D layout.
__global__ void __launch_bounds__(32)
jacobi_scan_kernel(const float* __restrict__ gm,
                   const float* __restrict__ gq,
                   float* __restrict__ gout) {
  const int b    = blockIdx.x;
  const int lane = threadIdx.x & 31;
  const int hf   = lane >> 4;   // lane half: selects K pair / row group
  const int lid  = lane & 15;

  const float* mb = gm   + (size_t)b * 256;
  const float* qb = gq   + (size_t)b * 256 * 3;
  float*       ob = gout + (size_t)b * 256 * 3;

  // ---- build A-layout operands (ISA 32-bit 16x4: lanes 0-15 K=0/1, 16-31 K=2/3)
  v2f Am[4], Ax[4], Ay[4], Az[4];   // channels: m, m*qx, m*qy, m*qz
  v2f U[4], L[4];                   // upper-tri ones / strict-lower ones (B / A roles)
  #pragma unroll
  for (int k = 0; k < 4; ++k) {
    #pragma unroll
    for (int j = 0; j < 2; ++j) {
      const int   kc = 4 * k + j + 2 * hf;   // column of X / row of U
      const int   e  = lid * 16 + kc;        // element index in batch
      const float mv = mb[e];
      const float q0 = qb[3 * e + 0];
      const float q1 = qb[3 * e + 1];
      const float q2 = qb[3 * e + 2];
      Am[k][j] = mv;
      Ax[k][j] = mv * q0;
      Ay[k][j] = mv * q1;
      Az[k][j] = mv * q2;
      U[k][j] = (kc <= lid) ? 1.0f : 0.0f;   // U[kc][n=lid]
      L[k][j] = (kc <  lid) ? 1.0f : 0.0f;   // Lstrict[m=lid][kc]
    }
  }

  // ---- pass 1: row-wise inclusive scans (K=16 via 4 chained K=4 WMMAs)
  v8f Sm = {}, Sx = {}, Sy = {}, Sz = {};
  #pragma unroll
  for (int k = 0; k < 4; ++k) {
    Sm = wmma4(Am[k], U[k], Sm);
    Sx = wmma4(Ax[k], U[k], Sx);
    Sy = wmma4(Ay[k], U[k], Sy);
    Sz = wmma4(Az[k], U[k], Sz);
  }

  // ---- pass 2: broadcast row totals, add strict-lower carries via WMMA
  auto carry = [&](v8f Y) -> v8f {
    float T[16];
    #pragma unroll
    for (int r = 0; r < 8; ++r) {
      T[r]     = __shfl(Y[r], 15, 32);   // Y[r][15]   (lane 15, vgpr r)
      T[r + 8] = __shfl(Y[r], 31, 32);   // Y[r+8][15] (lane 31, vgpr r)
    }
    v8f S = Y;                            // accumulate carries onto Y
    #pragma unroll
    for (int k = 0; k < 4; ++k) {
      v2f Bt;                             // B row kc broadcast across all N
      Bt[0] = hf ? T[4 * k + 2] : T[4 * k + 0];
      Bt[1] = hf ? T[4 * k + 3] : T[4 * k + 1];
      S = wmma4(L[k], Bt, S);
    }
    return S;
  };
  Sm = carry(Sm);
  Sx = carry(Sx);
  Sy = carry(Sy);
  Sz = carry(Sz);

  // ---- grand totals = inclusive scan at element 255 (lane 31, vgpr 7)
  const float Mtot = __shfl(Sm[7], 31, 32);
  const float Xt   = __shfl(Sx[7], 31, 32);
  const float Yt   = __shfl(Sy[7], 31, 32);
  const float Zt   = __shfl(Sz[7], 31, 32);
  const float rM   = 1.0f / Mtot;

  // ---- pointwise fixup + store; D layout: vgpr v, lane -> element (v+8*hf)*16+lid
  #pragma unroll
  for (int v = 0; v < 8; ++v) {
    const int   e  = (v + 8 * hf) * 16 + lid;
    const float mv = mb[e];
    const float q0 = qb[3 * e + 0];
    const float q1 = qb[3 * e + 1];
    const float q2 = qb[3 * e + 2];
    float r0, r1, r2;
    if (e == 0) {
      r0 = Xt * rM; r1 = Yt * rM; r2 = Zt * rM;
    } else {
      const float inv = 1.0f / (Sm[v] - mv);          // 1 / M[e-1]
      r0 = q0 - (Sx[v] - mv * q0) * inv;              // q - S_excl / M_prev
      r1 = q1 - (Sy[v] - mv * q1) * inv;
      r2 = q2 - (Sz[v] - mv * q2) * inv;
    }
    ob[3 * e + 0] = r0;
    ob[3 * e + 1] = r1;
    ob[3 * e + 2] = r2;
  }
}

extern "C" void kernel_launch(void* const* d_in, const int* in_sizes, int n_in,
                              void* d_out, int out_size, void* d_ws, size_t ws_size,
                              hipStream_t stream) {
  const float* m   = (const float*)d_in[0];   // (B, 256)
  const float* q   = (const float*)d_in[1];   // (B, 256, 3)
  float*       out = (float*)d_out;           // (B, 256, 3)
  const int B = in_sizes[0] / 256;
  hipLaunchKernelGGL(jacobi_scan_kernel, dim3(B), dim3(32), 0, stream, m, q, out);
}